// Exponential_71519795413075
// MI455X (gfx1250) — compile-verified
//
#include <hip/hip_runtime.h>
#include <stdint.h>

// Scaling-and-squaring diffeomorphic integration (VecInt), MI455X/gfx1250.
//
// flow0 = disp / 2^10;  repeat 10x: flow += trilinear_sample(flow, id + flow)
// Field: [C=3, D=128, H=160, W=160] f32 = 39.3 MB. Ping-pong buffers (d_ws,
// d_out) both fit in the 192MB L2 -> steps 2..10 run at L2 bandwidth.
// HBM floor is only ~80MB total; the kernel is gather/L2-bandwidth bound,
// so there is no matmul structure and WMMA is inapplicable.
//
// CDNA5 paths used (both directions of the Tensor Data Mover):
//  - tensor_load_to_lds:   each block's coalesced input (2 rows x 3 channels,
//    320 contiguous floats per channel, channel stride D*H*W) is DMA'd to LDS.
//  - tensor_store_from_lds: results staged in an LDS out-tile and written back
//    by one DMA per block, keeping the vector-memory pipe free for the 24
//    data-dependent trilinear gathers per voxel.
//  - s_wait_tensorcnt for completion, workgroup barriers for LDS visibility.

#define TS   10
#define Cc   3
#define Dd   128
#define Hh   160
#define Ww   160
#define HWc  (Hh * Ww)          // 25600
#define DHWc (Dd * Hh * Ww)     // 3276800
#define RPB  2                  // rows per block
#define TPB  (RPB * Ww)         // 320 threads = 10 wave32

typedef unsigned int u32x4 __attribute__((ext_vector_type(4)));
typedef int          i32x4 __attribute__((ext_vector_type(4)));
typedef int          i32x8 __attribute__((ext_vector_type(8)));

__device__ __forceinline__ int iclamp(int v, int lo, int hi) {
    return v < lo ? lo : (v > hi ? hi : v);
}

// Issue a TDM transfer of a [Cc x TPB] f32 tile (dim0 contiguous, dim1 stride
// DHWc elements) between global `ga` and LDS offset `lds`. is_load selects
// direction. Wave-uniform inputs; caller gates to a single wave.
__device__ __forceinline__ void tdm_tile(unsigned long long ga, unsigned lds,
                                         bool is_load)
{
    // D# group 0: count=1 | lds_addr | global_addr[56:0] | type=2
    u32x4 g0;
    g0[0] = 1u;
    g0[1] = lds;
    g0[2] = (unsigned)(ga & 0xFFFFFFFFull);
    g0[3] = ((unsigned)(ga >> 32) & 0x01FFFFFFu) | (2u << 30);

    // D# group 1: data_size=4B; tensor TPBx3; tile TPBx3; dim0_stride=DHW
    i32x8 g1;
    g1[0] = (int)(2u << 16);               // wg_mask=0, data_size=2 (4B)
    g1[1] = (int)((unsigned)TPB << 16);    // [63:48]  tensor_dim0 lo = 320
    g1[2] = (int)((unsigned)Cc << 16);     // [95:80]  tensor_dim1 lo = 3
    g1[3] = (int)((unsigned)TPB << 16);    // [127:112] tile_dim0 = 320
    g1[4] = (int)Cc;                       // [143:128] tile_dim1 = 3; tile_dim2 = 0
    g1[5] = (int)(unsigned)DHWc;           // [191:160] tensor_dim0_stride lo
    g1[6] = 0;
    g1[7] = 0;

    i32x4 gz; gz[0] = gz[1] = gz[2] = gz[3] = 0;   // groups 2/3 unused (2D tile)

#if __clang_major__ >= 23
    i32x8 gz8; gz8[0]=gz8[1]=gz8[2]=gz8[3]=gz8[4]=gz8[5]=gz8[6]=gz8[7]=0;
    if (is_load) __builtin_amdgcn_tensor_load_to_lds  (g0, g1, gz, gz, gz8, 0);
    else         __builtin_amdgcn_tensor_store_from_lds(g0, g1, gz, gz, gz8, 0);
#else
    if (is_load) __builtin_amdgcn_tensor_load_to_lds  (g0, g1, gz, gz, 0);
    else         __builtin_amdgcn_tensor_store_from_lds(g0, g1, gz, gz, 0);
#endif
}

__global__ __launch_bounds__(TPB) void vecint_step(const float* __restrict__ in,
                                                   float* __restrict__ out,
                                                   float scale)
{
    __shared__ float s_in [Cc * TPB];   // input tile  (3 ch x 320 voxels)
    __shared__ float s_out[Cc * TPB];   // output tile (3 ch x 320 voxels)

    const int pair = blockIdx.x;        // 0 .. D*H/2-1 (row pair, contiguous 320)
    const int tid  = threadIdx.x;       // 0 .. 319
    const int r    = (tid >= Ww) ? 1 : 0;
    const int x    = tid - r * Ww;
    const int z    = pair / (Hh / RPB);
    const int y    = (pair - z * (Hh / RPB)) * RPB + r;

    // ---- TDM load: stage the block's 2 rows x 3 channels into LDS ----
    if ((tid >> 5) == 0) {
        tdm_tile((unsigned long long)(uintptr_t)(in + (size_t)pair * TPB),
                 (unsigned)(uintptr_t)(&s_in[0]), /*is_load=*/true);
        __builtin_amdgcn_s_wait_tensorcnt(0);
    }
    __syncthreads();

    // ---- per-voxel flow (scaled) -> sample position in index space ----
    // align_corners=False: unnormalize(identity + f) == i + 0.5*Size*f
    const float v0 = s_in[0 * TPB + tid];
    const float v1 = s_in[1 * TPB + tid];
    const float v2 = s_in[2 * TPB + tid];

    const float sx = (float)x + 0.5f * (float)Ww * (v0 * scale);
    const float sy = (float)y + 0.5f * (float)Hh * (v1 * scale);
    const float sz = (float)z + 0.5f * (float)Dd * (v2 * scale);

    const float fxf = floorf(sx), fyf = floorf(sy), fzf = floorf(sz);
    const float tx = sx - fxf, ty = sy - fyf, tz = sz - fzf;
    const int   x0 = (int)fxf, y0 = (int)fyf, z0 = (int)fzf;

    // Per-axis taps: zero-padding folded into weights; indices clamped so the
    // (dead) loads stay in-bounds. Matches reference's padding_mode='zeros'.
    float wx[2], wy[2], wz[2];
    int   xi[2], yi[2], zi[2];

    wx[0] = (x0     >= 0 && x0     < Ww) ? (1.0f - tx) : 0.0f;
    wx[1] = (x0 + 1 >= 0 && x0 + 1 < Ww) ? tx          : 0.0f;
    xi[0] = iclamp(x0,     0, Ww - 1);
    xi[1] = iclamp(x0 + 1, 0, Ww - 1);

    wy[0] = (y0     >= 0 && y0     < Hh) ? (1.0f - ty) : 0.0f;
    wy[1] = (y0 + 1 >= 0 && y0 + 1 < Hh) ? ty          : 0.0f;
    yi[0] = iclamp(y0,     0, Hh - 1);
    yi[1] = iclamp(y0 + 1, 0, Hh - 1);

    wz[0] = (z0     >= 0 && z0     < Dd) ? (1.0f - tz) : 0.0f;
    wz[1] = (z0 + 1 >= 0 && z0 + 1 < Dd) ? tz          : 0.0f;
    zi[0] = iclamp(z0,     0, Dd - 1);
    zi[1] = iclamp(z0 + 1, 0, Dd - 1);

    // ---- 8-corner trilinear gather, 3 channels (linear in `in`, so scale
    //      factors out: out = scale * (in_voxel + sum w*in_corner)) ----
    float a0 = 0.0f, a1 = 0.0f, a2 = 0.0f;
#pragma unroll
    for (int c2 = 0; c2 < 2; ++c2) {
#pragma unroll
        for (int c1 = 0; c1 < 2; ++c1) {
            const int rb = (zi[c2] * Hh + yi[c1]) * Ww;
            const float wzy = wz[c2] * wy[c1];
#pragma unroll
            for (int c0 = 0; c0 < 2; ++c0) {
                const float w   = wzy * wx[c0];
                const int   idx = rb + xi[c0];
                a0 = fmaf(w, in[idx],            a0);
                a1 = fmaf(w, in[DHWc + idx],     a1);
                a2 = fmaf(w, in[2 * DHWc + idx], a2);
            }
        }
    }

    // ---- stage results in LDS, TDM store back to global ----
    s_out[0 * TPB + tid] = scale * (v0 + a0);
    s_out[1 * TPB + tid] = scale * (v1 + a1);
    s_out[2 * TPB + tid] = scale * (v2 + a2);
    __syncthreads();

    if ((tid >> 5) == 0) {
        tdm_tile((unsigned long long)(uintptr_t)(out + (size_t)pair * TPB),
                 (unsigned)(uintptr_t)(&s_out[0]), /*is_load=*/false);
        __builtin_amdgcn_s_wait_tensorcnt(0);  // hold LDS until DMA drains
    }
}

extern "C" void kernel_launch(void* const* d_in, const int* in_sizes, int n_in,
                              void* d_out, int out_size, void* d_ws, size_t ws_size,
                              hipStream_t stream)
{
    (void)in_sizes; (void)n_in; (void)out_size; (void)ws_size;

    const float* disp = (const float*)d_in[0];   // [3,128,160,160] f32
    // d_in[1] (identity grid) is unused: folded analytically into the kernel.

    float* out = (float*)d_out;                  // final result buffer
    float* tmp = (float*)d_ws;                   // 39.3 MB ping-pong buffer

    dim3 grid(Dd * Hh / RPB);   // 10240 blocks, one 2-row tile each
    dim3 block(TPB);            // 320 threads = 10 wave32

    // Step 1: disp -> tmp with scale = 2^-10 (division folded into the step).
    vecint_step<<<grid, block, 0, stream>>>(disp, tmp, 1.0f / 1024.0f);

    // Steps 2..10: ping-pong tmp <-> out; 9 flips starting at tmp end at out.
    float* bufs[2] = { tmp, out };
    int cur = 0;
    for (int t = 1; t < TS; ++t) {
        const int nxt = cur ^ 1;
        vecint_step<<<grid, block, 0, stream>>>(bufs[cur], bufs[nxt], 1.0f);
        cur = nxt;
    }
    // cur == 1 -> final flow is in d_out.
}